// EncodeStateDirectly_23888608100715
// MI455X (gfx1250) — compile-verified
//
#include <hip/hip_runtime.h>
#include <hip/hip_bf16.h>

typedef __attribute__((ext_vector_type(2))) float v2f;
typedef __attribute__((ext_vector_type(8))) float v8f;

// ---------------------------------------------------------------------------
// Register-blocked fp32 WMMA GEMM: D = (ACC ? D : 0) + opA(X) * opB(Y)
// One wave computes an (MR*16) x (NR*16) output block as MR x NR WMMA tiles,
// reusing each A fragment across NR tiles and each B fragment across MR tiles
// (raises L2 arithmetic intensity and WMMA issue density).
//   TA: A[m][k] = X[k*ldx + m]   (transposed read)
//   TB: B[k][n] = Y[n*ldy + k]   (transposed read)
//   NEGA: negate A fragment (rank-k downdate; fp32 WMMA has no NEG bits)
// ---------------------------------------------------------------------------
template <int MR, int NR, bool TA, bool TB, bool ACC, bool NEGA>
__global__ void wmma_gemm_f32(const float* __restrict__ X, int ldx,
                              const float* __restrict__ Y, int ldy,
                              float* __restrict__ D, int ldd,
                              int Mblocks, int Nblocks, int K) {
  int wid = (blockIdx.x * blockDim.x + threadIdx.x) >> 5;   // wave-uniform
  if (wid >= Mblocks * Nblocks) return;
  int mb = wid / Nblocks, nb = wid % Nblocks;
  int m0 = mb * (MR * 16), n0 = nb * (NR * 16);
  int lane = threadIdx.x & 31;
  int l    = lane & 15;          // column within tile for B/C/D, row for A
  int kk   = (lane >> 4) << 1;   // 0 or 2 : K sub-slot
  int rsel = (lane >> 4) << 3;   // 0 or 8 : C/D row base

  v8f c[MR][NR];
#pragma unroll
  for (int i = 0; i < MR; ++i)
#pragma unroll
    for (int j = 0; j < NR; ++j) {
      c[i][j] = {};
      if (ACC) {
#pragma unroll
        for (int r = 0; r < 8; ++r)
          c[i][j][r] = D[(m0 + i * 16 + rsel + r) * ldd + n0 + j * 16 + l];
      }
    }

#pragma unroll 2
  for (int k0 = 0; k0 < K; k0 += 4) {
    v2f af[MR], bf[NR];
#pragma unroll
    for (int i = 0; i < MR; ++i) {
      int mi = m0 + i * 16 + l;
      if (TA) {
        af[i].x = X[(k0 + kk) * ldx + mi];
        af[i].y = X[(k0 + kk + 1) * ldx + mi];
      } else {
        af[i].x = X[mi * ldx + k0 + kk];
        af[i].y = X[mi * ldx + k0 + kk + 1];
      }
      if (NEGA) { af[i].x = -af[i].x; af[i].y = -af[i].y; }
    }
#pragma unroll
    for (int j = 0; j < NR; ++j) {
      int nj = n0 + j * 16 + l;
      if (TB) {
        bf[j].x = Y[nj * ldy + k0 + kk];
        bf[j].y = Y[nj * ldy + k0 + kk + 1];
      } else {
        bf[j].x = Y[(k0 + kk) * ldy + nj];
        bf[j].y = Y[(k0 + kk + 1) * ldy + nj];
      }
    }
#pragma unroll
    for (int i = 0; i < MR; ++i)
#pragma unroll
      for (int j = 0; j < NR; ++j)
        c[i][j] = __builtin_amdgcn_wmma_f32_16x16x4_f32(
            false, af[i], false, bf[j], (short)0, c[i][j], false, false);
  }

#pragma unroll
  for (int i = 0; i < MR; ++i)
#pragma unroll
    for (int j = 0; j < NR; ++j)
#pragma unroll
      for (int r = 0; r < 8; ++r)
        D[(m0 + i * 16 + rsel + r) * ldd + n0 + j * 16 + l] = c[i][j][r];
}

// ---------------------------------------------------------------------------
// Residual pass: E = C(1024x1024) * v(1024x16); acc += sum |Yb_k - E|
// Yb_k[m][b] = x[b*32768 + k*1024 + m]  (xk = x + k*1024)
// ---------------------------------------------------------------------------
__global__ void wmma_resid(const float* __restrict__ Cm,
                           const float* __restrict__ v,
                           const float* __restrict__ xk,
                           float* __restrict__ acc) {
  int wid = (blockIdx.x * blockDim.x + threadIdx.x) >> 5;
  if (wid >= 64) return;
  int m0 = wid << 4;
  int lane = threadIdx.x & 31;
  int l = lane & 15, kk = (lane >> 4) << 1, rsel = (lane >> 4) << 3;

  v8f c = {};
#pragma unroll 4
  for (int k0 = 0; k0 < 1024; k0 += 4) {
    v2f a, b;
    a.x = Cm[(m0 + l) * 1024 + k0 + kk];
    a.y = Cm[(m0 + l) * 1024 + k0 + kk + 1];
    b.x = v[(k0 + kk) * 16 + l];
    b.y = v[(k0 + kk + 1) * 16 + l];
    c = __builtin_amdgcn_wmma_f32_16x16x4_f32(false, a, false, b, (short)0, c,
                                              false, false);
  }
  float s = 0.f;
#pragma unroll
  for (int r = 0; r < 8; ++r) {
    int row = m0 + rsel + r;
    float ref = xk[l * 32768 + row];
    s += fabsf(ref - c[r]);
  }
  for (int off = 16; off; off >>= 1) s += __shfl_xor(s, off, 32);
  if (lane == 0) atomicAdd(acc, s);
}

// ---------------------------------------------------------------------------
// 64x64 diagonal Cholesky block, in place at G[j0..j0+63][j0..j0+63] (lower).
// One workgroup of 64 threads (2 waves).
// ---------------------------------------------------------------------------
__global__ void potrf64(float* __restrict__ G, int j0) {
  __shared__ float S[64][65];
  int t = threadIdx.x;
  float* Gp = G + j0 * 1024 + j0;
  for (int cidx = 0; cidx < 64; ++cidx) S[t][cidx] = Gp[t * 1024 + cidx];
  __syncthreads();
  for (int cidx = 0; cidx < 64; ++cidx) {
    if (t == cidx) S[cidx][cidx] = sqrtf(S[cidx][cidx]);
    __syncthreads();
    if (t > cidx) S[t][cidx] /= S[cidx][cidx];
    __syncthreads();
    if (t > cidx) {
      float lrc = S[t][cidx];
      for (int cc = cidx + 1; cc <= t; ++cc) S[t][cc] -= lrc * S[cc][cidx];
    }
    __syncthreads();
  }
  for (int cidx = 0; cidx <= t; ++cidx) Gp[t * 1024 + cidx] = S[t][cidx];
}

// ---------------------------------------------------------------------------
// Panel TRSM: rows (j0+64 .. 1023) of columns j0..j0+63 solved against the
// lower-triangular 64x64 diagonal block. One thread per row.
// ---------------------------------------------------------------------------
__global__ void trsm64(float* __restrict__ G, int j0, int rem) {
  __shared__ float Ld[64][65];
  int t = threadIdx.x;
  for (int i = t; i < 64 * 64; i += blockDim.x) {
    int r = i >> 6, cidx = i & 63;
    Ld[r][cidx] = G[(j0 + r) * 1024 + (j0 + cidx)];
  }
  __syncthreads();
  int r = blockIdx.x * blockDim.x + t;
  if (r < rem) {
    float* row = G + (j0 + 64 + r) * 1024 + j0;
    float buf[64];
#pragma unroll 8
    for (int cidx = 0; cidx < 64; ++cidx) buf[cidx] = row[cidx];
    for (int cidx = 0; cidx < 64; ++cidx) {
      float v = buf[cidx];
      for (int k = 0; k < cidx; ++k) v -= buf[k] * Ld[cidx][k];
      buf[cidx] = v / Ld[cidx][cidx];
    }
#pragma unroll 8
    for (int cidx = 0; cidx < 64; ++cidx) row[cidx] = buf[cidx];
  }
}

// ---------------------------------------------------------------------------
// Forward substitution L z = rhs (1024 x 16 RHS). One wave per RHS column.
// ---------------------------------------------------------------------------
__global__ void trsv_fwd(const float* __restrict__ L,
                         const float* __restrict__ rhs,
                         float* __restrict__ z) {
  __shared__ float zc[1024];
  int b = blockIdx.x;
  int lane = threadIdx.x;
  for (int j = 0; j < 1024; ++j) {
    float s = 0.f;
    for (int k = lane; k < j; k += 32) s += L[j * 1024 + k] * zc[k];
    for (int off = 16; off; off >>= 1) s += __shfl_xor(s, off, 32);
    if (lane == 0) zc[j] = (rhs[j * 16 + b] - s) / L[j * 1024 + j];
    __syncthreads();
  }
  for (int j = lane; j < 1024; j += 32) z[j * 16 + b] = zc[j];
}

// Backward substitution L^T x = z. One wave per RHS column.
__global__ void trsv_bwd(const float* __restrict__ L,
                         const float* __restrict__ z,
                         float* __restrict__ x0) {
  __shared__ float xc[1024];
  int b = blockIdx.x;
  int lane = threadIdx.x;
  for (int j = 1023; j >= 0; --j) {
    float s = 0.f;
    for (int k = j + 1 + lane; k < 1024; k += 32) s += L[k * 1024 + j] * xc[k];
    for (int off = 16; off; off >>= 1) s += __shfl_xor(s, off, 32);
    if (lane == 0) xc[j] = (z[j * 16 + b] - s) / L[j * 1024 + j];
    __syncthreads();
  }
  for (int j = lane; j < 1024; j += 32) x0[j * 16 + b] = xc[j];
}

__global__ void zero_f32(float* __restrict__ p, int n) {
  int i = blockIdx.x * blockDim.x + threadIdx.x;
  if (i < n) p[i] = 0.f;
}

__global__ void finalize_loss(const float* __restrict__ acc,
                              float* __restrict__ out) {
  if (threadIdx.x == 0) out[0] = acc[0] * (1.0f / 524288.0f);  // mean over 32768*16
}

// ---------------------------------------------------------------------------
// Host-side orchestration (all launches on `stream`, graph-capture safe).
// Inputs: d_in[0]=step(int), d_in[1]=x(16*32*1024 f32), d_in[2]=y, d_in[3]=A,
// d_in[4]=C. Output: scalar loss.
// ---------------------------------------------------------------------------
extern "C" void kernel_launch(void* const* d_in, const int* in_sizes, int n_in,
                              void* d_out, int out_size, void* d_ws,
                              size_t ws_size, hipStream_t stream) {
  const float* x  = (const float*)d_in[1];
  const float* A  = (const float*)d_in[3];
  const float* Cm = (const float*)d_in[4];
  float* out = (float*)d_out;
  float* ws  = (float*)d_ws;

  // Workspace layout (floats)
  float* G   = ws;                    // 1024*1024  Gram / Cholesky factor
  float* rhs = G + 1024 * 1024;       // 1024*16    O^T Y^T
  float* acc = rhs + 1024 * 16;       // 16         |residual| accumulator
  float* zb  = acc + 16;              // 1024*16    intermediate solve
  float* T0  = zb + 1024 * 16;        // 1024*1024  Krylov chain ping
  float* T1  = T0 + 1024 * 1024;      // 1024*1024  Krylov chain pong
  float* v0  = T1 + 1024 * 1024;      // 1024*16    x0 / propagated state ping
  float* v1  = v0 + 1024 * 16;        // 1024*16    pong

  // Zero accumulated buffers (G, rhs, acc are contiguous)
  int zn = 1024 * 1024 + 1024 * 16 + 16;
  zero_f32<<<(zn + 255) / 256, 256, 0, stream>>>(ws, zn);

  // ---- Pass 1: G = sum_k T_k^T T_k ; rhs = sum_k T_k^T Yb_k ; T_{k+1}=T_k A
  // Big GEMMs: 2x2 register blocking -> 32x32 per wave, 1024 waves, 128 blocks.
  const float* Tcur = Cm;
  float* bufs[2] = {T0, T1};
  for (int k = 0; k < 32; ++k) {
    // G += T^T T   (1024x1024x1024, TRANS_A, accumulate)
    wmma_gemm_f32<2, 2, true, false, true, false><<<128, 256, 0, stream>>>(
        Tcur, 1024, Tcur, 1024, G, 1024, 32, 32, 1024);
    // rhs += T^T Yb_k   (Yb_k[m][b] = x[b*32768 + k*1024 + m] -> TRANS_B)
    // 4x1 blocking: one B fragment feeds 4 WMMAs. 16 waves -> 2 blocks.
    wmma_gemm_f32<4, 1, true, true, true, false><<<2, 256, 0, stream>>>(
        Tcur, 1024, x + (size_t)k * 1024, 32768, rhs, 16, 16, 1, 1024);
    if (k < 31) {
      float* Tn = bufs[k & 1];
      wmma_gemm_f32<2, 2, false, false, false, false><<<128, 256, 0, stream>>>(
          Tcur, 1024, A, 1024, Tn, 1024, 32, 32, 1024);
      Tcur = Tn;
    }
  }

  // ---- Blocked Cholesky: G = L L^T (lower triangle, in place)
  for (int j = 0; j < 1024; j += 64) {
    potrf64<<<1, 64, 0, stream>>>(G, j);
    int rem = 1024 - j - 64;
    if (rem > 0) {
      trsm64<<<(rem + 255) / 256, 256, 0, stream>>>(G, j, rem);
      // trailing -= L21 L21^T  (WMMA rank-64 downdate, A negated on load)
      int db = rem / 32;               // rem is a multiple of 64
      int waves = db * db;
      wmma_gemm_f32<2, 2, false, true, true, true>
          <<<(waves + 7) / 8, 256, 0, stream>>>(
              G + (j + 64) * 1024 + j, 1024, G + (j + 64) * 1024 + j, 1024,
              G + (j + 64) * 1024 + (j + 64), 1024, db, db, 64);
    }
  }

  // ---- Solve (L L^T) x0 = rhs   -> x0 stored in v0 as [1024][16]
  trsv_fwd<<<16, 32, 0, stream>>>(G, rhs, zb);
  trsv_bwd<<<16, 32, 0, stream>>>(G, zb, v0);

  // ---- Pass 2: Yhat_k = C v_k ; v_{k+1} = A v_k ; acc += sum |Yb_k - Yhat_k|
  float* vcur = v0;
  float* valt = v1;
  for (int k = 0; k < 32; ++k) {
    wmma_resid<<<8, 256, 0, stream>>>(Cm, vcur, x + (size_t)k * 1024, acc);
    if (k < 31) {
      wmma_gemm_f32<4, 1, false, false, false, false><<<2, 256, 0, stream>>>(
          A, 1024, vcur, 16, valt, 16, 16, 1, 1024);
      float* t = vcur; vcur = valt; valt = t;
    }
  }

  finalize_loss<<<1, 32, 0, stream>>>(acc, out);
}